// MP_network_1666447311389
// MI455X (gfx1250) — compile-verified
//
#include <hip/hip_runtime.h>
#include <hip/hip_bf16.h>

// ---------------------------------------------------------------------------
// MP-GNN on MI455X (gfx1250, wave32).  All GEMMs use V_WMMA_F32_16X16X4_F32
// (exact f32, matches reference numerics; workload is HBM-bound anyway).
// node_emb/agg (25.6 MB each) are L2-resident on the 192 MB L2, so the
// per-edge gather/scatter is cheap; only edge-major streams hit HBM.
// ---------------------------------------------------------------------------

typedef float v2f __attribute__((ext_vector_type(2)));
typedef float v8f __attribute__((ext_vector_type(8)));

#define N_NODES   50000
#define N_EDGES   500000
#define ATOM_DIM  128
#define BOND_DIM  64
#define EMB       128
#define N_GRAPHS  256

__device__ __forceinline__ v8f wmma_f32_4(v2f a, v2f b, v8f c) {
  // D = A(16x4 f32) * B(4x16 f32) + C(16x16 f32)
  return __builtin_amdgcn_wmma_f32_16x16x4_f32(
      /*neg_a=*/false, a, /*neg_b=*/false, b,
      /*c_mod=*/(short)0, c, /*reuse_a=*/false, /*reuse_b=*/false);
}

// out[m][n] = sum_k A[m][k] * W[n][k] + bias[n]
// One wave computes a 16x16 tile; blockIdx.x selects the 16-row tile,
// wave index selects the 16-col tile.  A/B lane layout per CDNA5 ISA 7.12.2.
__global__ void embed_gemm(const float* __restrict__ A,
                           const float* __restrict__ W,
                           const float* __restrict__ bias,
                           float* __restrict__ out,
                           int K, int Nout) {
  const int lane = threadIdx.x & 31;
  const int wave = threadIdx.x >> 5;
  const int half = lane >> 4;   // 0: K=0,1  1: K=2,3
  const int l15  = lane & 15;
  const int mbase = blockIdx.x * 16;
  const int nbase = wave * 16;
  if (nbase >= Nout) return;    // wave-uniform

  const float* arow = A + (size_t)(mbase + l15) * K;   // A row M=l15
  const float* wrow = W + (size_t)(nbase + l15) * K;   // B col N=l15 (= W row)

  v8f c = {};
  for (int kk = 0; kk < K; kk += 4) {
    const int kb = kk + 2 * half;
    v2f a = *(const v2f*)(arow + kb);
    v2f b = *(const v2f*)(wrow + kb);
    c = wmma_f32_4(a, b, c);
  }
  const float bv = bias[nbase + l15];
#pragma unroll
  for (int r = 0; r < 8; ++r) {
    const int m = mbase + r + 8 * half;
    out[(size_t)m * Nout + nbase + l15] = c[r] + bv;
  }
}

__global__ void zero_f32(float* __restrict__ p, unsigned n) {
  unsigned i = blockIdx.x * blockDim.x + threadIdx.x;
  if (i < n) p[i] = 0.0f;
}

// One wave per edge: msg = node_emb[src] * edge_emb; agg[dst] += msg (atomic).
// lane handles 4 consecutive f32 (128 = 32 lanes * float4).
__global__ void mp_scatter(const float* __restrict__ node_emb,
                           const float* __restrict__ edge_emb,
                           const int* __restrict__ src,
                           const int* __restrict__ dst,
                           float* __restrict__ agg, int nEdges) {
  const int e    = (int)((blockIdx.x * blockDim.x + threadIdx.x) >> 5);
  const int lane = threadIdx.x & 31;
  if (e >= nEdges) return;
  const int s = src[e];
  const int d = dst[e];
  const float4 nv = *(const float4*)(node_emb + (size_t)s * EMB + lane * 4);
  const float4 ev = *(const float4*)(edge_emb + (size_t)e * EMB + lane * 4);
  float* ap = agg + (size_t)d * EMB + lane * 4;
  atomicAdd(ap + 0, nv.x * ev.x);
  atomicAdd(ap + 1, nv.y * ev.y);
  atomicAdd(ap + 2, nv.z * ev.z);
  atomicAdd(ap + 3, nv.w * ev.w);
}

__global__ void vec_add_inplace(float* __restrict__ a,
                                const float* __restrict__ b, unsigned n) {
  unsigned i = blockIdx.x * blockDim.x + threadIdx.x;
  if (i < n) a[i] += b[i];
}

// Fused energy MLP + global-add-pool for a tile of 16 nodes.
// LDS rows padded to stride 130/66 so the A-operand column reads
// (fixed k, varying m) hit distinct banks.
__global__ void mlp_pool(const float* __restrict__ node_emb,
                         const float* __restrict__ W1, const float* __restrict__ b1,
                         const float* __restrict__ W2, const float* __restrict__ b2,
                         const float* __restrict__ W3,
                         const int* __restrict__ batch,
                         float* __restrict__ dg) {
  __shared__ float h [16 * 130];
  __shared__ float h1[16 * 130];
  __shared__ float h2[16 * 66];

  const int tid   = threadIdx.x;
  const int mbase = blockIdx.x * 16;

  // stage: relu(node_emb) tile -> LDS
  for (int i = tid; i < 16 * EMB; i += 256) {
    const int m = i >> 7, k = i & 127;
    h[m * 130 + k] = fmaxf(node_emb[(size_t)(mbase + m) * EMB + k], 0.0f);
  }
  __syncthreads();

  const int lane = tid & 31;
  const int wave = tid >> 5;
  const int half = lane >> 4;
  const int l15  = lane & 15;

  // layer 1: 128 -> 128, 8 waves x (16x16) tiles, K = 128
  {
    const int nbase = wave * 16;
    const int n = nbase + l15;
    v8f c = {};
    for (int kk = 0; kk < EMB; kk += 4) {
      const int kb = kk + 2 * half;
      v2f a; a.x = h[l15 * 130 + kb]; a.y = h[l15 * 130 + kb + 1];
      v2f b; b.x = W1[n * EMB + kb];  b.y = W1[n * EMB + kb + 1];
      c = wmma_f32_4(a, b, c);
    }
    const float bv = b1[n];
#pragma unroll
    for (int r = 0; r < 8; ++r) {
      const int m = r + 8 * half;
      h1[m * 130 + n] = fmaxf(c[r] + bv, 0.0f);
    }
  }
  __syncthreads();

  // layer 2: 128 -> 64, waves 0..3 (wave-uniform branch, EXEC stays full)
  if (wave < 4) {
    const int nbase = wave * 16;
    const int n = nbase + l15;
    v8f c = {};
    for (int kk = 0; kk < EMB; kk += 4) {
      const int kb = kk + 2 * half;
      v2f a; a.x = h1[l15 * 130 + kb];      a.y = h1[l15 * 130 + kb + 1];
      v2f b; b.x = W2[n * EMB + kb];        b.y = W2[n * EMB + kb + 1];
      c = wmma_f32_4(a, b, c);
    }
    const float bv = b2[n];
#pragma unroll
    for (int r = 0; r < 8; ++r) {
      const int m = r + 8 * half;
      h2[m * 66 + n] = fmaxf(c[r] + bv, 0.0f);
    }
  }
  __syncthreads();

  // layer 3: 64 -> 1 (no bias), then pool into graph bins
  if (tid < 16) {
    float e = 0.0f;
    for (int k = 0; k < 64; ++k) e += h2[tid * 66 + k] * W3[k];
    atomicAdd(&dg[batch[mbase + tid]], e);
  }
}

extern "C" void kernel_launch(void* const* d_in, const int* in_sizes, int n_in,
                              void* d_out, int out_size, void* d_ws, size_t ws_size,
                              hipStream_t stream) {
  const float* x         = (const float*)d_in[0];
  const float* edge_attr = (const float*)d_in[1];
  const int*   eidx      = (const int*)d_in[2];
  const int*   batch     = (const int*)d_in[3];
  const float* Wa        = (const float*)d_in[4];
  const float* ba        = (const float*)d_in[5];
  const float* Wb        = (const float*)d_in[6];
  const float* bb        = (const float*)d_in[7];
  const float* W1        = (const float*)d_in[8];
  const float* b1        = (const float*)d_in[9];
  const float* W2        = (const float*)d_in[10];
  const float* b2        = (const float*)d_in[11];
  const float* W3        = (const float*)d_in[12];
  float*       dg        = (float*)d_out;

  const int* src = eidx;
  const int* dst = eidx + N_EDGES;

  // workspace layout
  float* node_emb = (float*)d_ws;                            // 50000*128
  float* edge_emb = node_emb + (size_t)N_NODES * EMB;        // 500000*128
  float* agg      = edge_emb + (size_t)N_EDGES * EMB;        // 50000*128

  const unsigned NODE_F = N_NODES * EMB;                     // 6,400,000

  // Step 1: embeddings (WMMA f32 16x16x4)
  embed_gemm<<<N_NODES / 16, 256, 0, stream>>>(x, Wa, ba, node_emb, ATOM_DIM, EMB);
  embed_gemm<<<N_EDGES / 16, 256, 0, stream>>>(edge_attr, Wb, bb, edge_emb, BOND_DIM, EMB);

  // Step 2: two message-passing layers with residual
  for (int layer = 0; layer < 2; ++layer) {
    zero_f32<<<(NODE_F + 255) / 256, 256, 0, stream>>>(agg, NODE_F);
    mp_scatter<<<(N_EDGES * 32 + 255) / 256, 256, 0, stream>>>(
        node_emb, edge_emb, src, dst, agg, N_EDGES);
    vec_add_inplace<<<(NODE_F + 255) / 256, 256, 0, stream>>>(node_emb, agg, NODE_F);
  }

  // Step 3+4: energy MLP (WMMA) + global add pool
  zero_f32<<<1, 256, 0, stream>>>(dg, N_GRAPHS);
  mlp_pool<<<N_NODES / 16, 256, 0, stream>>>(node_emb, W1, b1, W2, b2, W3, batch, dg);
}